// GATNet_12489764897031
// MI455X (gfx1250) — compile-verified
//
#include <hip/hip_runtime.h>
#include <math.h>

#define N_NODES 50000
#define N_EDGES 800000
#define NE_TOT  (N_EDGES + N_NODES)   // self-loops appended
#define IN_FEATS 256
#define HID 64
#define HEADS 2
#define F1 (HEADS * HID)              // 128
#define NCLS 40

typedef __attribute__((ext_vector_type(2))) float v2f;
typedef __attribute__((ext_vector_type(8))) float v8f;

// ---------- order-preserving float<->uint map for atomicMax-based segment max ----------
static __device__ __forceinline__ unsigned fmap(float f) {
    unsigned u = __float_as_uint(f);
    return (u & 0x80000000u) ? ~u : (u | 0x80000000u);
}
static __device__ __forceinline__ float finv(unsigned t) {
    return (t & 0x80000000u) ? __uint_as_float(t & 0x7FFFFFFFu)
                             : __uint_as_float(~t);
}
static __device__ __forceinline__ void edge_sd(const int* __restrict__ ei, int e, int& s, int& d) {
    if (e < N_EDGES) { s = ei[e]; d = ei[N_EDGES + e]; }
    else             { s = e - N_EDGES; d = s; }           // self-loop
}
static __device__ __forceinline__ float lrelu(float x) { return x > 0.f ? x : 0.2f * x; }

// ============================================================================
// GEMM1: h1[N,128] = x[N,256] @ W1[256,128], fp32 WMMA 16x16x4.
// grid.x = 3125 M-tiles, 8 waves/block = 8 N-tiles of 16.
// ============================================================================
__global__ __launch_bounds__(256) void gat_gemm1_wmma(const float* __restrict__ x,
                                                      const float* __restrict__ W1,
                                                      float* __restrict__ h1) {
    const int mtile = blockIdx.x;              // 0..3124
    const int wave  = threadIdx.x >> 5;        // 0..7  -> N tile
    const int lane  = threadIdx.x & 31;
    const int half  = lane >> 4;               // 0/1 : K sub-phase & M-half for D
    const int l     = lane & 15;
    const int row   = mtile * 16 + l;          // A: M = lane&15 for both halves
    const int ncol  = wave * 16 + l;           // B/D: N = lane&15

    v8f acc = {};
    const float* xr = x + (size_t)row * IN_FEATS;
    for (int k = 0; k < IN_FEATS; k += 4) {
        // A 16x4: VGPR0 holds K=k(+2*half), VGPR1 holds K=k+1(+2*half)
        v2f a = *(const v2f*)(xr + k + 2 * half);
        // B 4x16: VGPR0 -> row K=k+2*half, VGPR1 -> row K=k+1+2*half, col=ncol
        const float* bp = W1 + (size_t)(k + 2 * half) * F1 + ncol;
        v2f b; b.x = bp[0]; b.y = bp[F1];
        acc = __builtin_amdgcn_wmma_f32_16x16x4_f32(false, a, false, b,
                                                    (short)0, acc, false, false);
    }
    // D 16x16 f32: VGPR r -> M = r + 8*half, N = l
    float* dr = h1 + (size_t)(mtile * 16 + 8 * half) * F1 + wave * 16 + l;
#pragma unroll
    for (int r = 0; r < 8; ++r) dr[(size_t)r * F1] = acc[r];
}

// ============================================================================
// GEMM2: h2[N,40] = h2in[N,128] @ W2[128,40]. 3 N-tiles (48 cols, mask >=40).
// ============================================================================
__global__ __launch_bounds__(96) void gat_gemm2_wmma(const float* __restrict__ h2in,
                                                     const float* __restrict__ W2,
                                                     float* __restrict__ h2) {
    const int mtile = blockIdx.x;
    const int wave  = threadIdx.x >> 5;        // 0..2
    const int lane  = threadIdx.x & 31;
    const int half  = lane >> 4;
    const int l     = lane & 15;
    const int row   = mtile * 16 + l;
    const int ncol  = wave * 16 + l;           // 0..47

    v8f acc = {};
    const float* xr = h2in + (size_t)row * F1;
    for (int k = 0; k < F1; k += 4) {
        v2f a = *(const v2f*)(xr + k + 2 * half);
        v2f b;
        if (ncol < NCLS) {
            const float* bp = W2 + (size_t)(k + 2 * half) * NCLS + ncol;
            b.x = bp[0]; b.y = bp[NCLS];
        } else { b.x = 0.f; b.y = 0.f; }
        acc = __builtin_amdgcn_wmma_f32_16x16x4_f32(false, a, false, b,
                                                    (short)0, acc, false, false);
    }
    if (ncol < NCLS) {
        float* dr = h2 + (size_t)(mtile * 16 + 8 * half) * NCLS + ncol;
#pragma unroll
        for (int r = 0; r < 8; ++r) dr[(size_t)r * NCLS] = acc[r];
    }
}

// ============================================================================
// Attention score dots
// ============================================================================
__global__ void gat_att1(const float* __restrict__ h1, const float* __restrict__ as1,
                         const float* __restrict__ ad1, float* __restrict__ a_src,
                         float* __restrict__ a_dst) {
    int i = blockIdx.x * blockDim.x + threadIdx.x;     // node*2 + head
    if (i >= N_NODES * HEADS) return;
    int n = i >> 1, h = i & 1;
    const float* hp = h1 + (size_t)n * F1 + h * HID;
    const float* s0 = as1 + h * HID;
    const float* d0 = ad1 + h * HID;
    float s = 0.f, d = 0.f;
#pragma unroll 8
    for (int c = 0; c < HID; ++c) { float v = hp[c]; s += v * s0[c]; d += v * d0[c]; }
    a_src[i] = s; a_dst[i] = d;
}

__global__ void gat_att2(const float* __restrict__ h2, const float* __restrict__ as2,
                         const float* __restrict__ ad2, float* __restrict__ a_src,
                         float* __restrict__ a_dst) {
    int n = blockIdx.x * blockDim.x + threadIdx.x;
    if (n >= N_NODES) return;
    const float* hp = h2 + (size_t)n * NCLS;
    float s = 0.f, d = 0.f;
#pragma unroll 8
    for (int c = 0; c < NCLS; ++c) { float v = hp[c]; s += v * as2[c]; d += v * ad2[c]; }
    a_src[n] = s; a_dst[n] = d;
}

// ============================================================================
// Edge pass 1: segment max via order-preserving uint atomicMax
// ============================================================================
__global__ void gat_edge_max1(const int* __restrict__ ei, const float* __restrict__ a_src,
                              const float* __restrict__ a_dst, unsigned* __restrict__ emax) {
    int e = blockIdx.x * blockDim.x + threadIdx.x;
    if (e >= NE_TOT) return;
    int s, d; edge_sd(ei, e, s, d);
#pragma unroll
    for (int h = 0; h < HEADS; ++h) {
        float x = lrelu(a_src[s * 2 + h] + a_dst[d * 2 + h]);
        atomicMax(&emax[d * 2 + h], fmap(x));
    }
}

__global__ void gat_edge_max2(const int* __restrict__ ei, const float* __restrict__ a_src,
                              const float* __restrict__ a_dst, unsigned* __restrict__ emax) {
    int e = blockIdx.x * blockDim.x + threadIdx.x;
    if (e >= NE_TOT) return;
    int s, d; edge_sd(ei, e, s, d);
    float x = lrelu(a_src[s] + a_dst[d]);
    atomicMax(&emax[d], fmap(x));
}

// ============================================================================
// Edge pass 2: p = exp(e - emax[dst]); denom[dst]+=p; accum[dst,:] += p*h[src,:]
// One wave32 per edge; each lane owns 4 contiguous feature columns (float4).
// ============================================================================
__global__ __launch_bounds__(256) void gat_edge_accum1(const int* __restrict__ ei,
        const float* __restrict__ a_src, const float* __restrict__ a_dst,
        const unsigned* __restrict__ emax, const float* __restrict__ h1,
        float* __restrict__ denom, float* __restrict__ accum) {
    int e = blockIdx.x * 8 + (threadIdx.x >> 5);
    if (e >= NE_TOT) return;
    const int lane = threadIdx.x & 31;
    int s, d; edge_sd(ei, e, s, d);
    float p0 = __expf(lrelu(a_src[s * 2 + 0] + a_dst[d * 2 + 0]) - finv(emax[d * 2 + 0]));
    float p1 = __expf(lrelu(a_src[s * 2 + 1] + a_dst[d * 2 + 1]) - finv(emax[d * 2 + 1]));
    if (lane == 0) { atomicAdd(&denom[d * 2 + 0], p0); atomicAdd(&denom[d * 2 + 1], p1); }
    const float4 hv = *(const float4*)(h1 + (size_t)s * F1 + lane * 4);
    const float p = (lane < 16) ? p0 : p1;        // cols < 64 -> head 0
    float* ap = accum + (size_t)d * F1 + lane * 4;
    atomicAdd(ap + 0, hv.x * p);
    atomicAdd(ap + 1, hv.y * p);
    atomicAdd(ap + 2, hv.z * p);
    atomicAdd(ap + 3, hv.w * p);
}

__global__ __launch_bounds__(256) void gat_edge_accum2(const int* __restrict__ ei,
        const float* __restrict__ a_src, const float* __restrict__ a_dst,
        const unsigned* __restrict__ emax, const float* __restrict__ h2,
        float* __restrict__ denom, float* __restrict__ accum) {
    int e = blockIdx.x * 8 + (threadIdx.x >> 5);
    if (e >= NE_TOT) return;
    const int lane = threadIdx.x & 31;
    int s, d; edge_sd(ei, e, s, d);
    float p = __expf(lrelu(a_src[s] + a_dst[d]) - finv(emax[d]));
    if (lane == 0) atomicAdd(&denom[d], p);
    for (int c = lane; c < NCLS; c += 32)
        atomicAdd(&accum[(size_t)d * NCLS + c], h2[(size_t)s * NCLS + c] * p);
}

// ============================================================================
// Node finalize: divide by denom, add bias, activation
// ============================================================================
__global__ void gat_finalize1(const float* __restrict__ accum, const float* __restrict__ denom,
                              const float* __restrict__ b1, float* __restrict__ h2in) {
    int i = blockIdx.x * blockDim.x + threadIdx.x;
    if (i >= N_NODES * F1) return;
    int n = i >> 7, c = i & 127;
    float v = accum[i] / denom[n * 2 + (c >> 6)] + b1[c];
    h2in[i] = v > 0.f ? v : (expf(v) - 1.0f);             // ELU
}

__global__ void gat_finalize2(const float* __restrict__ accum, const float* __restrict__ denom,
                              const float* __restrict__ b2, float* __restrict__ out) {
    int i = blockIdx.x * blockDim.x + threadIdx.x;
    if (i >= N_NODES * NCLS) return;
    int n = i / NCLS, c = i - n * NCLS;
    out[i] = accum[i] / denom[n] + b2[c];
}

// ============================================================================
extern "C" void kernel_launch(void* const* d_in, const int* in_sizes, int n_in,
                              void* d_out, int out_size, void* d_ws, size_t ws_size,
                              hipStream_t stream) {
    const float* x    = (const float*)d_in[0];
    const int*   ei   = (const int*)  d_in[1];
    const float* W1   = (const float*)d_in[2];
    const float* as1  = (const float*)d_in[3];
    const float* ad1  = (const float*)d_in[4];
    const float* b1   = (const float*)d_in[5];
    const float* W2   = (const float*)d_in[6];
    const float* as2  = (const float*)d_in[7];
    const float* ad2  = (const float*)d_in[8];
    const float* b2   = (const float*)d_in[9];
    float* out = (float*)d_out;

    // ---- workspace layout (floats). Zero-init block is contiguous. ----
    float* ws      = (float*)d_ws;
    float* h1      = ws;                              // N*128 (reused as h2in after finalize1)
    float* accum1  = h1     + (size_t)N_NODES * F1;   // N*128  } zero
    float* accum2  = accum1 + (size_t)N_NODES * F1;   // N*40   } zero
    float* denom1  = accum2 + (size_t)N_NODES * NCLS; // N*2    } zero
    float* denom2  = denom1 + (size_t)N_NODES * 2;    // N      } zero
    unsigned* emax1 = (unsigned*)(denom2 + N_NODES);  // N*2    } zero (below fmap(-inf), min sentinel)
    unsigned* emax2 = emax1 + (size_t)N_NODES * 2;    // N      } zero
    float* a_src1  = (float*)(emax2 + N_NODES);       // N*2
    float* a_dst1  = a_src1 + (size_t)N_NODES * 2;    // N*2
    float* a_src2  = a_dst1 + (size_t)N_NODES * 2;    // N
    float* a_dst2  = a_src2 + N_NODES;                // N
    float* h2      = a_dst2 + N_NODES;                // N*40

    const size_t zeroFloats = (size_t)N_NODES * (F1 + NCLS + 2 + 1 + 2 + 1); // accum1..emax2
    hipMemsetAsync(accum1, 0, zeroFloats * sizeof(float), stream);

    const int MT = N_NODES / 16;                      // 3125 exact
    // ---- Layer 1 ----
    gat_gemm1_wmma<<<MT, 256, 0, stream>>>(x, W1, h1);
    gat_att1<<<(N_NODES * HEADS + 255) / 256, 256, 0, stream>>>(h1, as1, ad1, a_src1, a_dst1);
    gat_edge_max1<<<(NE_TOT + 255) / 256, 256, 0, stream>>>(ei, a_src1, a_dst1, emax1);
    gat_edge_accum1<<<(NE_TOT + 7) / 8, 256, 0, stream>>>(ei, a_src1, a_dst1, emax1, h1,
                                                          denom1, accum1);
    gat_finalize1<<<(N_NODES * F1 + 255) / 256, 256, 0, stream>>>(accum1, denom1, b1, h1);
    // ---- Layer 2 (h1 now holds ELU'd layer-1 output) ----
    gat_gemm2_wmma<<<MT, 96, 0, stream>>>(h1, W2, h2);
    gat_att2<<<(N_NODES + 255) / 256, 256, 0, stream>>>(h2, as2, ad2, a_src2, a_dst2);
    gat_edge_max2<<<(NE_TOT + 255) / 256, 256, 0, stream>>>(ei, a_src2, a_dst2, emax2);
    gat_edge_accum2<<<(NE_TOT + 7) / 8, 256, 0, stream>>>(ei, a_src2, a_dst2, emax2, h2,
                                                          denom2, accum2);
    gat_finalize2<<<(N_NODES * NCLS + 255) / 256, 256, 0, stream>>>(accum2, denom2, b2, out);
}